// ClusterNetHomogeneous_74947179315776
// MI455X (gfx1250) — compile-verified
//
#include <hip/hip_runtime.h>
#include <hip/hip_bf16.h>

typedef float v2f __attribute__((ext_vector_type(2)));
typedef float v8f __attribute__((ext_vector_type(8)));

#define N_NODES 100000
#define N_EDGES 1600000
#define N_GRAPHS 64
#define HDIM 128
#define OUTDIM 4

// ---------------------------------------------------------------------------
// z = h  (vectorized float4 copy; seeds the GIN self-term before scatter-add)
// ---------------------------------------------------------------------------
__global__ void gin_copy_f4(const float* __restrict__ in, float* __restrict__ out, long n4) {
    long i = (long)blockIdx.x * blockDim.x + threadIdx.x;
    if (i < n4) ((float4*)out)[i] = ((const float4*)in)[i];
}

// ---------------------------------------------------------------------------
// z[dst] += h[src]   (per-edge, per-float4-chunk; f32 atomics hit L2 — the
// whole 51MB feature matrix is L2-resident on MI455X's 192MB L2)
// ---------------------------------------------------------------------------
__global__ void gin_scatter_add(const float* __restrict__ H, const int* __restrict__ src,
                                const int* __restrict__ dst, float* __restrict__ Z,
                                long total, int d4) {
    long tid = (long)blockIdx.x * blockDim.x + threadIdx.x;
    if (tid >= total) return;
    int e = (int)(tid / d4);
    int c = (int)(tid % d4);
    int s = src[e];
    int t = dst[e];
    float4 v = ((const float4*)H)[(size_t)s * d4 + c];
    float* zp = Z + ((size_t)t * d4 + c) * 4;
    atomicAdd(zp + 0, v.x);
    atomicAdd(zp + 1, v.y);
    atomicAdd(zp + 2, v.z);
    atomicAdd(zp + 3, v.w);
}

// ---------------------------------------------------------------------------
// Out[N x 128] = relu( (Z[N x K] @ W[K x 128] + b) * (gamma/sqrt(1+eps)) + beta )
// fp32 WMMA 16x16x4. One wave32 -> one 16-row tile x all 8 column tiles.
// W staged in LDS (<=64KB), reused by 8 waves per block.
// ---------------------------------------------------------------------------
__global__ void __launch_bounds__(256)
gin_mlp_wmma(const float* __restrict__ Z, const float* __restrict__ W,
             const float* __restrict__ bias, const float* __restrict__ gamma,
             const float* __restrict__ beta, float* __restrict__ Out,
             int N, int K) {
    __shared__ float sW[HDIM * HDIM];  // K*128 floats used, K <= 128 (64KB max)

    for (int i = threadIdx.x; i < K * HDIM; i += blockDim.x) sW[i] = W[i];
    __syncthreads();

    const int lane = threadIdx.x & 31;
    const int wave = threadIdx.x >> 5;
    const int tile = blockIdx.x * 8 + wave;          // 16-row tile index
    const int n_tiles = (N + 15) >> 4;
    if (tile >= n_tiles) return;                     // wave-uniform: EXEC stays all-1s for WMMA

    const int row_base = tile * 16;
    const int half = lane >> 4;                      // lane half selects K pair / M-offset 8
    const int r = lane & 15;

    v8f acc[8];
    const v8f vzero = {0.f, 0.f, 0.f, 0.f, 0.f, 0.f, 0.f, 0.f};
#pragma unroll
    for (int c = 0; c < 8; ++c) acc[c] = vzero;

    for (int k = 0; k < K; k += 4) {
        const int ka = k + 2 * half;                 // fp32 A layout: V0 = K,K+2; V1 = K+1,K+3
        v2f a;
        const float* zrow = Z + (size_t)(row_base + r) * K + ka;
        a.x = zrow[0];
        a.y = zrow[1];
#pragma unroll
        for (int c = 0; c < 8; ++c) {
            v2f b;
            b.x = sW[ka * HDIM + c * 16 + r];        // B row striped across lanes
            b.y = sW[(ka + 1) * HDIM + c * 16 + r];
            acc[c] = __builtin_amdgcn_wmma_f32_16x16x4_f32(
                false, a, false, b, (short)0, acc[c], false, false);
        }
    }

    const float inv = rsqrtf(1.0f + 1e-5f);          // eval-mode BN with running var = 1
#pragma unroll
    for (int c = 0; c < 8; ++c) {
        const int col = c * 16 + r;
        const float sc = gamma[col] * inv;
        const float bs = beta[col];
        const float bi = bias[col];
#pragma unroll
        for (int v = 0; v < 8; ++v) {
            const int row = row_base + v + 8 * half; // fp32 C/D layout
            float y = (acc[c][v] + bi) * sc + bs;
            Out[(size_t)row * HDIM + col] = fmaxf(y, 0.0f);
        }
    }
}

// ---------------------------------------------------------------------------
// pooled init to 0 (valid identity: post-ReLU features are >= 0)
// ---------------------------------------------------------------------------
__global__ void gin_zero(float* __restrict__ p, int n) {
    int i = blockIdx.x * blockDim.x + threadIdx.x;
    if (i < n) p[i] = 0.0f;
}

// ---------------------------------------------------------------------------
// segment max: pooled[batch[n]][f] = max over nodes (int-bit atomicMax, exact
// for non-negative floats)
// ---------------------------------------------------------------------------
__global__ void gin_pool_max(const float* __restrict__ H, const int* __restrict__ batch,
                             float* __restrict__ P, int N) {
    long tid = (long)blockIdx.x * blockDim.x + threadIdx.x;
    if (tid >= (long)N * HDIM) return;
    int n = (int)(tid >> 7);
    int f = (int)(tid & (HDIM - 1));
    int g = batch[n];
    float v = H[tid];
    atomicMax((int*)&P[(size_t)g * HDIM + f], __float_as_int(v));
}

// ---------------------------------------------------------------------------
// out[64 x 4] = pooled[64 x 128] @ Wout[128 x 4] + bout  (tiny, one block)
// ---------------------------------------------------------------------------
__global__ void gin_out_proj(const float* __restrict__ P, const float* __restrict__ Wout,
                             const float* __restrict__ bout, float* __restrict__ out) {
    int tid = threadIdx.x;                 // 256 = 64 graphs * 4 outputs
    int g = tid >> 2;
    int o = tid & 3;
    float s = bout[o];
    for (int k = 0; k < HDIM; ++k) s += P[g * HDIM + k] * Wout[k * OUTDIM + o];
    out[g * OUTDIM + o] = s;
}

extern "C" void kernel_launch(void* const* d_in, const int* in_sizes, int n_in,
                              void* d_out, int out_size, void* d_ws, size_t ws_size,
                              hipStream_t stream) {
    const float* x     = (const float*)d_in[0];
    const int*   ei    = (const int*)d_in[1];
    const int*   batch = (const int*)d_in[2];
    const int N = in_sizes[2];
    const int E = in_sizes[1] / 2;
    const int* src = ei;        // edge_index[0]
    const int* dst = ei + E;    // edge_index[1]

    // params flat in setup_inputs dict order:
    // for l in 0..2, j in 0..1: W, b, gamma, beta  (indices 3..26), then Wout, bout
    auto Pf = [&](int i) { return (const float*)d_in[i]; };

    float* b0 = (float*)d_ws;                       // z (aggregated input)
    float* b1 = b0 + (size_t)N * HDIM;              // t (hidden of MLP)
    float* b2 = b1 + (size_t)N * HDIM;              // h (layer output)
    float* pooled = b2 + (size_t)N * HDIM;          // 64 x 128

    const int n_tiles = (N + 15) / 16;
    const int mlp_blocks = (n_tiles + 7) / 8;

    const float* h_in = x;
    int din = 64;
    for (int l = 0; l < 3; ++l) {
        const int base = 3 + l * 8;
        // z = h_in ; z[dst] += h_in[src]
        long n4 = (long)N * din / 4;
        gin_copy_f4<<<(int)((n4 + 255) / 256), 256, 0, stream>>>(h_in, b0, n4);
        long st = (long)E * (din / 4);
        gin_scatter_add<<<(int)((st + 255) / 256), 256, 0, stream>>>(h_in, src, dst, b0, st, din / 4);
        // t = relu(bn(z @ W0 + b0))
        gin_mlp_wmma<<<mlp_blocks, 256, 0, stream>>>(b0, Pf(base + 0), Pf(base + 1),
                                                     Pf(base + 2), Pf(base + 3), b1, N, din);
        // h = relu(bn(t @ W1 + b1))
        gin_mlp_wmma<<<mlp_blocks, 256, 0, stream>>>(b1, Pf(base + 4), Pf(base + 5),
                                                     Pf(base + 6), Pf(base + 7), b2, N, HDIM);
        h_in = b2;
        din = HDIM;
    }

    // global max pool + output projection
    gin_zero<<<(N_GRAPHS * HDIM + 255) / 256, 256, 0, stream>>>(pooled, N_GRAPHS * HDIM);
    long pt = (long)N * HDIM;
    gin_pool_max<<<(int)((pt + 255) / 256), 256, 0, stream>>>(b2, batch, pooled, N);
    gin_out_proj<<<1, N_GRAPHS * OUTDIM, 0, stream>>>(pooled, Pf(27), Pf(28), (float*)d_out);
}